// SKD_64424509440533
// MI455X (gfx1250) — compile-verified
//
#include <hip/hip_runtime.h>

typedef __attribute__((ext_vector_type(2))) float v2f;
typedef __attribute__((ext_vector_type(8))) float v8f;

// ---------------------------------------------------------------------------
// Kernel 1: channel mean.  f: [b, C, HW] -> v: [b, HW],  v[b,p] = mean_ch f[b,ch,p]
// Compile-time C/HW -> constant strides, unrolled reduction.
// ---------------------------------------------------------------------------
template <int C, int HW>
__global__ __launch_bounds__(256)
void chan_mean_kernel(const float* __restrict__ f, float* __restrict__ v, int total) {
  int idx = blockIdx.x * blockDim.x + threadIdx.x;
  if (idx >= total) return;
  int b = idx / HW;            // HW is a power of two -> shift
  int p = idx - b * HW;
  const float* base = f + (size_t)b * (size_t)(C * HW) + p;
  float s = 0.0f;
#pragma unroll 8
  for (int ch = 0; ch < C; ++ch) s += base[(size_t)ch * HW];
  v[idx] = s * (1.0f / (float)C);
}

// ---------------------------------------------------------------------------
// Kernel 2: per-batch scale.  scale[b] = (sum v^2 - (sum v)^2/HW) / (HW - 1)
// One block (256 threads) per batch row.
// ---------------------------------------------------------------------------
template <int HW>
__global__ __launch_bounds__(256)
void row_scale_kernel(const float* __restrict__ v, float* __restrict__ scale) {
  __shared__ float s1[256];
  __shared__ float s2[256];
  int b = blockIdx.x;
  const float* row = v + (size_t)b * HW;
  float a = 0.0f, q = 0.0f;
#pragma unroll
  for (int i = threadIdx.x; i < HW; i += 256) {
    float x = row[i];
    a += x;
    q += x * x;
  }
  s1[threadIdx.x] = a;
  s2[threadIdx.x] = q;
  __syncthreads();
  for (int off = 128; off > 0; off >>= 1) {
    if (threadIdx.x < off) {
      s1[threadIdx.x] += s1[threadIdx.x + off];
      s2[threadIdx.x] += s2[threadIdx.x + off];
    }
    __syncthreads();
  }
  if (threadIdx.x == 0) {
    float sum = s1[0], sumsq = s2[0];
    float S = sumsq - sum * sum * (1.0f / (float)HW);
    scale[b] = S * (1.0f / (float)(HW - 1));
  }
}

// ---------------------------------------------------------------------------
// Kernel 3: scaled outer product via WMMA rank-1 GEMM.
// One wave32 computes one 16x16 f32 tile:
//   D = A(16x4) x B(4x16),  A col K=0 = scale * v[i0..i0+15], B row K=0 = v[k0..k0+15]
// A layout (32-bit, 16x4): lanes 0-15 VGPR0 = K=0 (M=lane); lanes 16-31 = K=2 -> zeroed.
// C/D layout: VGPR r, lane l -> row r + 8*(l>=16), col l&15.
// HW compile-time -> row strides become immediate store offsets (max 7*HW*4 < 2^23).
// ---------------------------------------------------------------------------
template <int HW>
__global__ __launch_bounds__(256)
void outer_wmma_kernel(const float* __restrict__ v, const float* __restrict__ scale,
                       float* __restrict__ out, long total_waves) {
  constexpr int NT = HW / 16;           // tiles per side (64 or 16)
  constexpr int TILES = NT * NT;        // 4096 or 256 -> power of two

  int lane = threadIdx.x & 31;
  int wid  = threadIdx.x >> 5;
  long wg  = (long)blockIdx.x * 8 + wid;
  if (wg >= total_waves) return;        // wave-uniform: EXEC all-ones past here

  int batch = (int)(wg >> __builtin_ctz(TILES));     // wg / TILES (pow2)
  int t     = (int)(wg & (TILES - 1));
  int ti    = t >> __builtin_ctz(NT);                // t / NT (pow2)
  int tk    = t & (NT - 1);
  int i0    = ti * 16;
  int k0    = tk * 16;

  const float* row = v + (size_t)batch * HW;
  float sc = scale[batch];

  v2f a  = {0.0f, 0.0f};
  v2f bm = {0.0f, 0.0f};
  if (lane < 16) {
    a.x  = sc * row[i0 + lane];   // A column K=0 (scale folded into A => exact rank-1)
    bm.x = row[k0 + lane];        // B row K=0
  }

  v8f c = {};
  c = __builtin_amdgcn_wmma_f32_16x16x4_f32(
      /*neg_a=*/false, a, /*neg_b=*/false, bm,
      /*c_mod=*/(short)0, c, /*reuse_a=*/false, /*reuse_b=*/false);

  // Single per-lane base address; 8 stores with compile-time immediate offsets.
  int rowAdd = (lane >= 16) ? 8 : 0;
  int col    = lane & 15;
  float* obase = out + (size_t)batch * (size_t)(HW * HW)
                     + (size_t)(i0 + rowAdd) * HW + (size_t)k0 + col;
#pragma unroll
  for (int r = 0; r < 8; ++r) {
    obase[r * HW] = c[r];               // r*HW*4 folds into IOFFSET
  }
}

// ---------------------------------------------------------------------------
// Launcher
// ---------------------------------------------------------------------------
template <int C, int HW>
static void run_one(const float* f, float* v, float* sc, float* out, hipStream_t stream) {
  constexpr int B = 32;
  // 1) channel means
  int total = B * HW;
  chan_mean_kernel<C, HW><<<(total + 255) / 256, 256, 0, stream>>>(f, v, total);
  // 2) per-row scale
  row_scale_kernel<HW><<<B, 256, 0, stream>>>(v, sc);
  // 3) scaled outer product, one WMMA per 16x16 tile, 8 waves per block
  constexpr long total_waves = (long)B * (long)(HW / 16) * (long)(HW / 16);
  constexpr long blocks = (total_waves + 7) / 8;
  outer_wmma_kernel<HW><<<(int)blocks, 256, 0, stream>>>(v, sc, out, total_waves);
}

extern "C" void kernel_launch(void* const* d_in, const int* in_sizes, int n_in,
                              void* d_out, int out_size, void* d_ws, size_t ws_size,
                              hipStream_t stream) {
  const float* ft0 = (const float*)d_in[0];  // [32, 64, 32, 32]
  const float* ft1 = (const float*)d_in[1];  // [32, 128, 16, 16]
  const float* fs0 = (const float*)d_in[2];  // [32, 32, 32, 32]
  const float* fs1 = (const float*)d_in[3];  // [32, 64, 16, 16]
  // d_in[4] = error_index: unused by the reference output.

  float* ws  = (float*)d_ws;
  float* out = (float*)d_out;

  // Workspace layout (floats): v arrays then scales (total ~328 KB).
  float* v0 = ws;            // 32*1024
  float* v1 = ws + 32768;    // 32*256
  float* v2 = ws + 40960;    // 32*1024
  float* v3 = ws + 73728;    // 32*256
  float* sc = ws + 81920;    // 4*32

  // Output offsets (tuple order: cov_t0, cov_t1, cov_s0, cov_s1), flat f32.
  const size_t o1 = (size_t)32 * 1024 * 1024;
  const size_t o2 = o1 + (size_t)32 * 256 * 256;
  const size_t o3 = o2 + (size_t)32 * 1024 * 1024;

  run_one<64, 1024>(ft0, v0, sc + 0,  out,      stream);
  run_one<128, 256>(ft1, v1, sc + 32, out + o1, stream);
  run_one<32, 1024>(fs0, v2, sc + 64, out + o2, stream);
  run_one<64, 256>( fs1, v3, sc + 96, out + o3, stream);
}